// Model_2628519985208
// MI455X (gfx1250) — compile-verified
//
#include <hip/hip_runtime.h>
#include <math.h>

#define NN 192
#define MM (192 * 192)
#define KSPLIT 8
#define CHUNK (MM / KSPLIT)  /* 4608 samples per adj K-chunk */
#define NTILES ((NN / 16) * (NN / 16)) /* 144 */
#define TWO_PI_F 6.283185307179586f

typedef __attribute__((ext_vector_type(2))) float v2f;
typedef __attribute__((ext_vector_type(8))) float v8f;

typedef __attribute__((address_space(1))) int gint_t;  // global int
typedef __attribute__((address_space(3))) int lint_t;  // LDS int

__device__ __forceinline__ v8f wmma4(v2f a, v2f b, v8f c) {
  // D(16x16,f32) = A(16x4,f32) x B(4x16,f32) + C
  return __builtin_amdgcn_wmma_f32_16x16x4_f32(
      /*neg_a=*/false, a, /*neg_b=*/false, b,
      /*c_mod=*/(short)0, c, /*reuse_a=*/false, /*reuse_b=*/false);
}

// ---------------------------------------------------------------------------
// Precompute A = exp(-2pi i kx p), B = exp(-2pi i ky p) as planar re/im.
// ---------------------------------------------------------------------------
__global__ void mats_kernel(const float* __restrict__ traj,
                            float* __restrict__ Ar, float* __restrict__ Ai,
                            float* __restrict__ Br, float* __restrict__ Bi) {
  int idx = blockIdx.x * blockDim.x + threadIdx.x;
  if (idx >= MM * NN) return;
  int m = idx / NN;
  int x = idx - m * NN;
  float p = (float)x - 96.0f;  // arange(N) - N/2
  float kx = traj[2 * m], ky = traj[2 * m + 1];
  float s, c;
  __sincosf(-TWO_PI_F * kx * p, &s, &c);
  Ar[idx] = c; Ai[idx] = s;
  __sincosf(-TWO_PI_F * ky * p, &s, &c);
  Br[idx] = c; Bi[idx] = s;
}

__global__ void init_kernel(float* __restrict__ w, float* __restrict__ dr,
                            float* __restrict__ di) {
  int m = blockIdx.x * blockDim.x + threadIdx.x;
  if (m < MM) { w[m] = 1.0f; dr[m] = 1.0f; di[m] = 0.0f; }
}

// ---------------------------------------------------------------------------
// adjoint (K-split): partial[x,y] = sum_{m in chunk} conj(A[m,x])*d[m]*conj(B[m,y])
// grid (12,12,KSPLIT), one wave per block. d chunk staged into LDS via the
// gfx1250 async global->LDS path when available; fragments read as ds b64.
// Re += Ur*Br + Ui*Bi ; Im += Ui*Br - Ur*Bi   (U = conj(A)*d)
// ---------------------------------------------------------------------------
template <bool PRE>
__global__ __launch_bounds__(32) void adj_kernel(
    const float* __restrict__ Ar, const float* __restrict__ Ai,
    const float* __restrict__ Br, const float* __restrict__ Bi,
    const float* __restrict__ traj,
    const float* __restrict__ dr, const float* __restrict__ di,
    float* __restrict__ partR, float* __restrict__ partI) {
  __shared__ float sdr[CHUNK];
  __shared__ float sdi[CHUNK];

  int x0 = blockIdx.x * 16, y0 = blockIdx.y * 16;
  int mBase = blockIdx.z * CHUNK;
  int lane = threadIdx.x;
  int r = lane & 15;   // A-op row / B-op col
  int hi = lane >> 4;  // half-wave select (K pair)

  // ---- stage d chunk into LDS ----
#if __has_builtin(__builtin_amdgcn_global_load_async_to_lds_b32)
  for (int i = lane; i < CHUNK; i += 32) {
    __builtin_amdgcn_global_load_async_to_lds_b32(
        (gint_t*)(dr + mBase + i), (lint_t*)(&sdr[i]), 0, 0);
    __builtin_amdgcn_global_load_async_to_lds_b32(
        (gint_t*)(di + mBase + i), (lint_t*)(&sdi[i]), 0, 0);
  }
#if __has_builtin(__builtin_amdgcn_s_wait_asynccnt)
  __builtin_amdgcn_s_wait_asynccnt(0);
#else
  asm volatile("s_wait_asynccnt 0" ::: "memory");
#endif
#else
  // fallback: plain copy (single wave; LDS ops are in-order within a wave)
  for (int i = lane; i < CHUNK; i += 32) {
    sdr[i] = dr[mBase + i];
    sdi[i] = di[mBase + i];
  }
#endif

  float px = (float)(x0 + r) - 96.0f;
  float py = (float)(y0 + r) - 96.0f;

  v8f accR = {};
  v8f accI = {};
  for (int lm = 0; lm < CHUNK; lm += 4) {
    v2f aUr, aUi, bR, bI;
    v2f d2r = *(const v2f*)&sdr[lm + 2 * hi];  // ds_load_b64
    v2f d2i = *(const v2f*)&sdi[lm + 2 * hi];
#pragma unroll
    for (int j = 0; j < 2; ++j) {
      int gm = mBase + lm + hi * 2 + j;
      float ar, ai, br, bi2;
      if (PRE) {
        int ia = gm * NN + x0 + r;
        int ib = gm * NN + y0 + r;
        ar = Ar[ia]; ai = Ai[ia];
        br = Br[ib]; bi2 = Bi[ib];
      } else {
        float kx = traj[2 * gm], ky = traj[2 * gm + 1];
        __sincosf(-TWO_PI_F * kx * px, &ai, &ar);
        __sincosf(-TWO_PI_F * ky * py, &bi2, &br);
      }
      float dre = d2r[j], dim2 = d2i[j];
      aUr[j] = ar * dre + ai * dim2;  // Re(conj(A)*d)
      aUi[j] = ar * dim2 - ai * dre;  // Im(conj(A)*d)
      bR[j] = br; bI[j] = bi2;
    }
    v2f aUrN = -aUr;  // f32 WMMA NEG bits must be 0 -> negate in VALU
    accR = wmma4(aUr, bR, accR);
    accR = wmma4(aUi, bI, accR);
    accI = wmma4(aUi, bR, accI);
    accI = wmma4(aUrN, bI, accI);
  }

  int tile = blockIdx.x * (NN / 16) + blockIdx.y;  // 0..143
  float* pR = partR + ((size_t)blockIdx.z * NTILES + tile) * 256;
  float* pI = partI + ((size_t)blockIdx.z * NTILES + tile) * 256;
#pragma unroll
  for (int rr = 0; rr < 8; ++rr) {
    int e = (rr + 8 * hi) * 16 + r;  // C/D layout: row = rr + 8*(lane/16)
    pR[e] = accR[rr];
    pI[e] = accI[rr];
  }
}

// deterministic fixed-order chunk reduction: partials -> image
__global__ void reduce_kernel(const float* __restrict__ partR,
                              const float* __restrict__ partI,
                              float* __restrict__ outR, float* __restrict__ outI) {
  int e = blockIdx.x * blockDim.x + threadIdx.x;  // 0 .. NTILES*256-1
  if (e >= NTILES * 256) return;
  int tile = e >> 8, within = e & 255;
  float sr = 0.0f, si = 0.0f;
#pragma unroll
  for (int c = 0; c < KSPLIT; ++c) {
    sr += partR[((size_t)c * NTILES + tile) * 256 + within];
    si += partI[((size_t)c * NTILES + tile) * 256 + within];
  }
  int row = within >> 4, col = within & 15;
  int x = (tile / (NN / 16)) * 16 + row;
  int y = (tile % (NN / 16)) * 16 + col;
  outR[x * NN + y] = sr;
  outI[x * NN + y] = si;
}

// ---------------------------------------------------------------------------
// forward op: t[m,x] = sum_y B[m,y]*img[x,y]; k[m] = sum_x A[m,x]*t[m,x]
// One wave per 16 rows of M. GEMM tiles over x, then row-dot with A via
// per-lane FMA + shfl_xor reduction inside each 16-lane half.
// ---------------------------------------------------------------------------
template <bool PRE>
__global__ __launch_bounds__(32) void op_kernel(
    const float* __restrict__ Ar, const float* __restrict__ Ai,
    const float* __restrict__ Br, const float* __restrict__ Bi,
    const float* __restrict__ traj,
    const float* __restrict__ imgR, const float* __restrict__ imgI,
    float* __restrict__ kr, float* __restrict__ ki) {
  int m0 = blockIdx.x * 16;
  int lane = threadIdx.x;
  int r = lane & 15;
  int hi = lane >> 4;

  float kyv = 0.0f;
  float kxv[8];
  if (!PRE) {
    kyv = traj[2 * (m0 + r) + 1];
#pragma unroll
    for (int rr = 0; rr < 8; ++rr) kxv[rr] = traj[2 * (m0 + rr + 8 * hi)];
  }

  float pr[8], pi2[8];
#pragma unroll
  for (int rr = 0; rr < 8; ++rr) { pr[rr] = 0.0f; pi2[rr] = 0.0f; }

  const float* browR = PRE ? (Br + (m0 + r) * NN) : nullptr;
  const float* browI = PRE ? (Bi + (m0 + r) * NN) : nullptr;

  for (int xt = 0; xt < NN / 16; ++xt) {
    int x0 = xt * 16;
    const float* irow = imgR + (x0 + r) * NN;
    const float* iirow = imgI + (x0 + r) * NN;
    v8f tR = {};
    v8f tI = {};
    for (int y = 0; y < NN; y += 4) {
      int gy0 = y + hi * 2;  // even -> 8B-aligned pair loads
      v2f aR, aI;
      if (PRE) {
        aR = *(const v2f*)(browR + gy0);  // global_load_b64
        aI = *(const v2f*)(browI + gy0);
      } else {
#pragma unroll
        for (int j = 0; j < 2; ++j) {
          float br, bi2;
          __sincosf(-TWO_PI_F * kyv * ((float)(gy0 + j) - 96.0f), &bi2, &br);
          aR[j] = br; aI[j] = bi2;
        }
      }
      v2f gR = *(const v2f*)(irow + gy0);   // img[x, y] pair (stride-NN rows)
      v2f gI = *(const v2f*)(iirow + gy0);
      v2f aIN = -aI;
      tR = wmma4(aR, gR, tR);
      tR = wmma4(aIN, gI, tR);
      tI = wmma4(aR, gI, tI);
      tI = wmma4(aI, gR, tI);
    }
    // row-dot of this t tile with A[m, x0..x0+15]
#pragma unroll
    for (int rr = 0; rr < 8; ++rr) {
      int gm = m0 + rr + 8 * hi;
      int gx = x0 + r;
      float ar, ai;
      if (PRE) {
        int ia = gm * NN + gx;
        ar = Ar[ia]; ai = Ai[ia];
      } else {
        __sincosf(-TWO_PI_F * kxv[rr] * ((float)gx - 96.0f), &ai, &ar);
      }
      float tr = tR[rr], ti = tI[rr];
      pr[rr]  += ar * tr - ai * ti;
      pi2[rr] += ar * ti + ai * tr;
    }
  }
  // reduce across the 16 lanes of each half (xor masks stay inside the half)
#pragma unroll
  for (int rr = 0; rr < 8; ++rr) {
    float sr = pr[rr], si = pi2[rr];
#pragma unroll
    for (int off = 8; off > 0; off >>= 1) {
      sr += __shfl_xor(sr, off, 32);
      si += __shfl_xor(si, off, 32);
    }
    if (r == 0) {
      int gm = m0 + rr + 8 * hi;
      kr[gm] = sr;
      ki[gm] = si;
    }
  }
}

__global__ void wdiv_kernel(float* __restrict__ w, const float* __restrict__ qr,
                            const float* __restrict__ qi,
                            float* __restrict__ dr, float* __restrict__ di) {
  int m = blockIdx.x * blockDim.x + threadIdx.x;
  if (m < MM) {
    float mag = sqrtf(qr[m] * qr[m] + qi[m] * qi[m]);
    float wn = w[m] / fmaxf(mag, 1e-20f);
    w[m] = wn;
    dr[m] = wn;   // next adjoint input (w is real)
    di[m] = 0.0f;
  }
}

__global__ void scale_kernel(const float* __restrict__ w,
                             const float* __restrict__ kr,
                             const float* __restrict__ ki,
                             float* __restrict__ dr, float* __restrict__ di) {
  int m = blockIdx.x * blockDim.x + threadIdx.x;
  if (m < MM) {
    dr[m] = w[m] * kr[m];
    di[m] = w[m] * ki[m];
  }
}

extern "C" void kernel_launch(void* const* d_in, const int* in_sizes, int n_in,
                              void* d_out, int out_size, void* d_ws, size_t ws_size,
                              hipStream_t stream) {
  (void)in_sizes; (void)n_in; (void)out_size;
  const float* x    = (const float*)d_in[0];  // (2, N, N) f32
  const float* traj = (const float*)d_in[1];  // (M, 2)    f32
  float* out = (float*)d_out;                 // (2, N, N) f32
  float* ws = (float*)d_ws;

  float* w     = ws;
  float* dr    = ws + 1 * MM;
  float* di    = ws + 2 * MM;
  float* qr    = ws + 3 * MM;
  float* qi    = ws + 4 * MM;
  float* imR   = ws + 5 * MM;
  float* imI   = ws + 6 * MM;
  float* kr    = ws + 7 * MM;
  float* ki    = ws + 8 * MM;
  float* partR = ws + 9 * MM;    // KSPLIT*NTILES*256 = 8*MM floats
  float* partI = ws + 17 * MM;   // another 8*MM floats

  const size_t matElems = (size_t)MM * NN;  // per matrix plane
  bool pre = ws_size >= (25ull * MM + 4ull * matElems) * sizeof(float);
  float *Ar = nullptr, *Ai = nullptr, *Br = nullptr, *Bi = nullptr;
  if (pre) {
    Ar = ws + 25 * MM;
    Ai = Ar + matElems;
    Br = Ai + matElems;
    Bi = Br + matElems;
    int total = MM * NN;
    mats_kernel<<<(total + 255) / 256, 256, 0, stream>>>(traj, Ar, Ai, Br, Bi);
  }

  init_kernel<<<(MM + 255) / 256, 256, 0, stream>>>(w, dr, di);

  dim3 adjGrid(NN / 16, NN / 16, KSPLIT);
  dim3 redGrid((NTILES * 256 + 255) / 256);
  dim3 opGrid(MM / 16);

  for (int it = 0; it < 3; ++it) {
    if (pre) {
      adj_kernel<true><<<adjGrid, 32, 0, stream>>>(Ar, Ai, Br, Bi, traj, dr, di, partR, partI);
    } else {
      adj_kernel<false><<<adjGrid, 32, 0, stream>>>(Ar, Ai, Br, Bi, traj, dr, di, partR, partI);
    }
    reduce_kernel<<<redGrid, 256, 0, stream>>>(partR, partI, imR, imI);
    if (pre) {
      op_kernel<true><<<opGrid, 32, 0, stream>>>(Ar, Ai, Br, Bi, traj, imR, imI, qr, qi);
    } else {
      op_kernel<false><<<opGrid, 32, 0, stream>>>(Ar, Ai, Br, Bi, traj, imR, imI, qr, qi);
    }
    wdiv_kernel<<<(MM + 255) / 256, 256, 0, stream>>>(w, qr, qi, dr, di);
  }

  // final forward NUFFT of the input image, density-compensate, adjoint to out
  if (pre) {
    op_kernel<true><<<opGrid, 32, 0, stream>>>(Ar, Ai, Br, Bi, traj, x, x + NN * NN, kr, ki);
  } else {
    op_kernel<false><<<opGrid, 32, 0, stream>>>(Ar, Ai, Br, Bi, traj, x, x + NN * NN, kr, ki);
  }
  scale_kernel<<<(MM + 255) / 256, 256, 0, stream>>>(w, kr, ki, dr, di);
  if (pre) {
    adj_kernel<true><<<adjGrid, 32, 0, stream>>>(Ar, Ai, Br, Bi, traj, dr, di, partR, partI);
  } else {
    adj_kernel<false><<<adjGrid, 32, 0, stream>>>(Ar, Ai, Br, Bi, traj, dr, di, partR, partI);
  }
  reduce_kernel<<<redGrid, 256, 0, stream>>>(partR, partI, out, out + NN * NN);
}